// RKNTransitionCell_29008209117325
// MI455X (gfx1250) — compile-verified
//
#include <hip/hip_runtime.h>
#include <math.h>

typedef float v2f __attribute__((ext_vector_type(2)));
typedef float v8f __attribute__((ext_vector_type(8)));

#define NB   8192
#define DD   64
#define OO   32
#define KMIX 15
#define RR   3
#define NHID 60
#define LOG2PI 1.8378770664093453f

// ---- optional gfx1250 async global->LDS path (ASYNCcnt-tracked DMA) ----
#if defined(__HIP_DEVICE_COMPILE__) && \
    __has_builtin(__builtin_amdgcn_global_load_async_to_lds_b128) && \
    __has_builtin(__builtin_amdgcn_s_wait_asynccnt)
#define USE_ASYNC_LDS 1
typedef int v4i __attribute__((ext_vector_type(4)));
typedef v4i __attribute__((address_space(1))) as1_v4i;
typedef v4i __attribute__((address_space(3))) as3_v4i;
static __device__ __forceinline__ void async_ld_b128(const void* g, void* l) {
  __builtin_amdgcn_global_load_async_to_lds_b128((as1_v4i*)g, (as3_v4i*)l, 0, 0);
}
#else
#define USE_ASYNC_LDS 0
#endif

// ---- 16x16 f32 C-tile GEMM helper on V_WMMA_F32_16X16X4_F32 ----
// A is MxK (row-major, lda) unless TA (then A(m,k)=Am[k*lda+m]).
// B is KxN (row-major, ldb) unless TB (then B(k,n)=Bm[n*ldb+k]).
// ISA layouts (CDNA5 ISA):
//   A(m,k): lane = m + 16*(k>>1), vgpr = k&1
//   B(k,n): lane = n + 16*(k>>1), vgpr = k&1
//   C(m,n): lane = n + 16*(m>=8), vgpr = m&7
template<bool TA, bool TB>
static __device__ __forceinline__
v8f wmma_tile(const float* Am, int lda, const float* Bm, int ldb,
              int mBase, int nBase, int kTot, int lane) {
  v8f acc = {0.f, 0.f, 0.f, 0.f, 0.f, 0.f, 0.f, 0.f};
  const int mh = lane & 15;
  const int kh = lane >> 4;
  for (int k0 = 0; k0 < kTot; k0 += 4) {
    const int ka = k0 + 2 * kh;
    v2f a, b;
    if constexpr (TA) {
      a.x = Am[ka * lda + mBase + mh];
      a.y = Am[(ka + 1) * lda + mBase + mh];
    } else {
      a.x = Am[(mBase + mh) * lda + ka];
      a.y = Am[(mBase + mh) * lda + ka + 1];
    }
    if constexpr (TB) {
      b.x = Bm[(nBase + mh) * ldb + ka];
      b.y = Bm[(nBase + mh) * ldb + ka + 1];
    } else {
      b.x = Bm[ka * ldb + nBase + mh];
      b.y = Bm[(ka + 1) * ldb + nBase + mh];
    }
    acc = __builtin_amdgcn_wmma_f32_16x16x4_f32(false, a, false, b,
                                                (short)0, acc, false, false);
  }
  return acc;
}

static __device__ __forceinline__
void store_tile(float* Cm, int ldc, int mBase, int nBase, v8f c, int lane) {
  const int mh = lane & 15;
  const int hi = lane >> 4;
#pragma unroll
  for (int v = 0; v < 8; ++v)
    Cm[(mBase + v + 8 * hi) * ldc + nBase + mh] = c[v];
}

__global__ __launch_bounds__(128)
void rkn_cell_kernel(const float* __restrict__ gMean, const float* __restrict__ gCov,
                     const float* __restrict__ gObs,
                     const float* __restrict__ gW1, const float* __restrict__ gB1,
                     const float* __restrict__ gW2, const float* __restrict__ gB2,
                     const float* __restrict__ gBasis,
                     const float* __restrict__ gPnf, const float* __restrict__ gPnd,
                     const float* __restrict__ gOnf, const float* __restrict__ gOnd,
                     const float* __restrict__ gHw, const float* __restrict__ gHb,
                     float* __restrict__ oMean, float* __restrict__ oCov,
                     float* __restrict__ oLL) {
  // LDS: 62.5 KB, heavily overlapped across stages.
  __shared__ float sF[4096];   // F;  later: PHt in [0..2047], Kg in [2048..4095] (ld 32)
  __shared__ float sP[4096];   // state_cov P;  later pred_cov PC
  __shared__ float sT[4096];   // T1 = F*P;  later Linv in [0..1023];  later KH
  __shared__ float sS[1024];   // S -> L (in place) -> S_inv
  __shared__ float sHw[2048];  // Hw (32x64)
  __shared__ float sMean[64], sPM[64], sH[64], sWts[16], sInnov[32], sRed[32];
  __shared__ float sLd;

  const int tid  = threadIdx.x;
  const int wave = tid >> 5;
  const int lane = tid & 31;
  const int b    = blockIdx.x;

  // ---- stage 0: stage P, Hw, mean into LDS ----
#if USE_ASYNC_LDS
  for (int i = tid; i < 1024; i += 128)
    async_ld_b128(gCov + (size_t)b * 4096 + i * 4, &sP[i * 4]);
  for (int i = tid; i < 512; i += 128)
    async_ld_b128(gHw + i * 4, &sHw[i * 4]);
#else
  {
    const float4* src = (const float4*)(gCov + (size_t)b * 4096);
    float4* dst = (float4*)sP;
    for (int i = tid; i < 1024; i += 128) dst[i] = src[i];
    const float4* hsrc = (const float4*)gHw;
    float4* hdst = (float4*)sHw;
    for (int i = tid; i < 512; i += 128) hdst[i] = hsrc[i];
  }
#endif
  if (tid < 64) sMean[tid] = gMean[(size_t)b * 64 + tid];
#if USE_ASYNC_LDS
  __builtin_amdgcn_s_wait_asynccnt(0);
#endif
  __syncthreads();

  // ---- MLP gate: h = elu(W1 m + b1) ----
  if (tid < NHID) {
    float acc = gB1[tid];
    for (int d = 0; d < 64; ++d) acc += gW1[tid * 64 + d] * sMean[d];
    sH[tid] = acc > 0.f ? acc : expm1f(acc);
  }
  __syncthreads();
  if (tid < KMIX) {
    float acc = gB2[tid];
    for (int h = 0; h < NHID; ++h) acc += gW2[tid * NHID + h] * sH[h];
    sWts[tid] = acc;  // logits
  }
  __syncthreads();
  if (tid == 0) {  // softmax over 15
    float mx = sWts[0];
    for (int k = 1; k < KMIX; ++k) mx = fmaxf(mx, sWts[k]);
    float sum = 0.f;
    for (int k = 0; k < KMIX; ++k) { float e = expf(sWts[k] - mx); sWts[k] = e; sum += e; }
    float inv = 1.f / sum;
    for (int k = 0; k < KMIX; ++k) sWts[k] *= inv;
  }
  __syncthreads();

  // ---- F = sum_k w_k basis_k  (basis is L2-resident; b128 loads) ----
  {
    const float4* bs4 = (const float4*)gBasis;
    float4* f4 = (float4*)sF;
    for (int e = tid; e < 1024; e += 128) {
      float4 acc = make_float4(0.f, 0.f, 0.f, 0.f);
#pragma unroll
      for (int k = 0; k < KMIX; ++k) {
        const float w = sWts[k];
        const float4 v = bs4[k * 1024 + e];
        acc.x += w * v.x; acc.y += w * v.y; acc.z += w * v.z; acc.w += w * v.w;
      }
      f4[e] = acc;
    }
  }
  __syncthreads();

  // ---- pred_mean = F m ----
  if (tid < 64) {
    float acc = 0.f;
    for (int j = 0; j < 64; ++j) acc += sF[tid * 64 + j] * sMean[j];
    sPM[tid] = acc;
  }
  __syncthreads();

  // ---- T1 = F * P  (64x64x64, WMMA) ----
  for (int nt = 0; nt < 4; ++nt) {
    v8f c = wmma_tile<false, false>(sF, 64, sP, 64, wave * 16, nt * 16, 64, lane);
    store_tile(sT, 64, wave * 16, nt * 16, c, lane);
  }
  // innovation = obs - (Hw pred_mean + Hb)   (divergent tail; re-converges before WMMA)
  if (tid < OO) {
    float acc = gHb[tid];
    for (int d = 0; d < 64; ++d) acc += sHw[tid * 64 + d] * sPM[d];
    sInnov[tid] = gObs[(size_t)b * OO + tid] - acc;
  }
  __syncthreads();

  // ---- PC = T1 * F^T + Q  -> sP  (Q = pnf pnf^T + diag(exp pnd), fused epilogue) ----
  for (int nt = 0; nt < 4; ++nt) {
    v8f c = wmma_tile<false, true>(sT, 64, sF, 64, wave * 16, nt * 16, 64, lane);
    const int mh = lane & 15, hi = lane >> 4;
#pragma unroll
    for (int v = 0; v < 8; ++v) {
      int m = wave * 16 + v + 8 * hi, n = nt * 16 + mh;
      float q = 0.f;
#pragma unroll
      for (int r = 0; r < RR; ++r) q += gPnf[m * RR + r] * gPnf[n * RR + r];
      if (m == n) q += expf(gPnd[m]);
      sP[m * 64 + n] = c[v] + q;
    }
  }
  __syncthreads();

  // ---- PHt = PC * Hw^T  (64x64x32) -> sF[0..2047] (ld 32) ----
  for (int nt = 0; nt < 2; ++nt) {
    v8f c = wmma_tile<false, true>(sP, 64, sHw, 64, wave * 16, nt * 16, 64, lane);
    store_tile(sF, 32, wave * 16, nt * 16, c, lane);
  }
  __syncthreads();

  // ---- S = Hw * PHt + Rn  (32x32) -> sS ; zero Linv area ----
  {
    const int mt = wave >> 1, nt = wave & 1;
    v8f c = wmma_tile<false, false>(sHw, 64, sF, 32, mt * 16, nt * 16, 64, lane);
    const int mh = lane & 15, hi = lane >> 4;
#pragma unroll
    for (int v = 0; v < 8; ++v) {
      int m = mt * 16 + v + 8 * hi, n = nt * 16 + mh;
      float r0 = 0.f;
#pragma unroll
      for (int r = 0; r < RR; ++r) r0 += gOnf[m * RR + r] * gOnf[n * RR + r];
      if (m == n) r0 += expf(gOnd[m]);
      sS[m * 32 + n] = c[v] + r0;
    }
  }
  for (int i = tid; i < 1024; i += 128) sT[i] = 0.f;  // T1 dead; becomes Linv
  __syncthreads();

  // ---- Cholesky: sS lower triangle -> L (in place) ----
  for (int j = 0; j < 32; ++j) {
    float s = 0.f;
    if (tid >= j && tid < 32) {
      s = sS[tid * 32 + j];
      for (int k = 0; k < j; ++k) s -= sS[tid * 32 + k] * sS[j * 32 + k];
    }
    __syncthreads();
    if (tid == j) sS[j * 32 + j] = sqrtf(fmaxf(s, 1e-30f));
    __syncthreads();
    if (tid > j && tid < 32) sS[tid * 32 + j] = s / sS[j * 32 + j];
    __syncthreads();
  }

  // ---- logdet ; Linv = L^{-1} (lane j owns column j; deps are lane-local) ----
  if (tid == 0) {
    float ld = 0.f;
    for (int j = 0; j < 32; ++j) ld += logf(sS[j * 32 + j]);
    sLd = 2.f * ld;
  }
  if (tid < 32) {
    const int j = tid;
    sT[j * 32 + j] = 1.f / sS[j * 32 + j];
    for (int i = j + 1; i < 32; ++i) {
      float acc = 0.f;
      for (int k = j; k < i; ++k) acc += sS[i * 32 + k] * sT[k * 32 + j];
      sT[i * 32 + j] = -acc / sS[i * 32 + i];
    }
  }
  __syncthreads();

  // ---- S_inv = Linv^T * Linv  (32x32x32) -> sS ----
  {
    const int mt = wave >> 1, nt = wave & 1;
    v8f c = wmma_tile<true, false>(sT, 32, sT, 32, mt * 16, nt * 16, 32, lane);
    store_tile(sS, 32, mt * 16, nt * 16, c, lane);
  }
  __syncthreads();

  // ---- Kg = PHt * S_inv  (64x32x32) -> sF[2048..] ; quad partials ----
  for (int nt = 0; nt < 2; ++nt) {
    v8f c = wmma_tile<false, false>(sF, 32, sS, 32, wave * 16, nt * 16, 32, lane);
    store_tile(sF + 2048, 32, wave * 16, nt * 16, c, lane);
  }
  if (tid < 32) {
    float t = 0.f;
    for (int p = 0; p < 32; ++p) t += sS[tid * 32 + p] * sInnov[p];
    sRed[tid] = sInnov[tid] * t;
  }
  __syncthreads();

  // ---- KH = Kg * Hw  (64x32x64) -> sT ; post_mean ----
  for (int nt = 0; nt < 4; ++nt) {
    v8f c = wmma_tile<false, false>(sF + 2048, 32, sHw, 64, wave * 16, nt * 16, 32, lane);
    store_tile(sT, 64, wave * 16, nt * 16, c, lane);
  }
  if (tid < 64) {
    float acc = sPM[tid];
    for (int o = 0; o < 32; ++o) acc += sF[2048 + tid * 32 + o] * sInnov[o];
    oMean[(size_t)b * 64 + tid] = acc;
  }
  __syncthreads();

  // ---- post_cov = PC - KH*PC  -> global ; log_likelihood ----
  float* ocov = oCov + (size_t)b * 4096;
  for (int nt = 0; nt < 4; ++nt) {
    v8f c = wmma_tile<false, false>(sT, 64, sP, 64, wave * 16, nt * 16, 64, lane);
    const int mh = lane & 15, hi = lane >> 4;
#pragma unroll
    for (int v = 0; v < 8; ++v) {
      int m = wave * 16 + v + 8 * hi, n = nt * 16 + mh;
      ocov[m * 64 + n] = sP[m * 64 + n] - c[v];
    }
  }
  if (tid == 0) {
    float quad = 0.f;
    for (int o = 0; o < 32; ++o) quad += sRed[o];
    oLL[b] = -0.5f * (sLd + quad + (float)OO * LOG2PI);
  }
}

extern "C" void kernel_launch(void* const* d_in, const int* in_sizes, int n_in,
                              void* d_out, int out_size, void* d_ws, size_t ws_size,
                              hipStream_t stream) {
  (void)in_sizes; (void)n_in; (void)out_size; (void)d_ws; (void)ws_size;
  const float* gMean  = (const float*)d_in[0];
  const float* gCov   = (const float*)d_in[1];
  const float* gObs   = (const float*)d_in[2];
  const float* gW1    = (const float*)d_in[3];
  const float* gB1    = (const float*)d_in[4];
  const float* gW2    = (const float*)d_in[5];
  const float* gB2    = (const float*)d_in[6];
  const float* gBasis = (const float*)d_in[7];
  const float* gPnf   = (const float*)d_in[8];
  const float* gPnd   = (const float*)d_in[9];
  const float* gOnf   = (const float*)d_in[10];
  const float* gOnd   = (const float*)d_in[11];
  const float* gHw    = (const float*)d_in[12];
  const float* gHb    = (const float*)d_in[13];

  float* out   = (float*)d_out;
  float* oMean = out;                                          // [B, D]
  float* oCov  = out + (size_t)NB * DD;                        // [B, D, D]
  float* oLL   = out + (size_t)NB * DD + (size_t)NB * DD * DD; // [B]

  rkn_cell_kernel<<<dim3(NB), dim3(128), 0, stream>>>(
      gMean, gCov, gObs, gW1, gB1, gW2, gB2, gBasis,
      gPnf, gPnd, gOnf, gOnd, gHw, gHb, oMean, oCov, oLL);
}